// SparseToDense_85177791414440
// MI455X (gfx1250) — compile-verified
//
#include <hip/hip_runtime.h>
#include <hip/hip_bf16.h>
#include <stdint.h>

// Problem constants (match reference)
constexpr int Hc   = 480;
constexpr int Wc   = 640;
constexpr int DIMc = 32;
constexpr int Bc   = 8;
constexpr long long HWc    = (long long)Hc * Wc;       // 307200
constexpr long long SITESc = (long long)Bc * HWc;      // 2,457,600
constexpr long long OUTNc  = SITESc * DIMc;            // 78,643,200 floats
constexpr int SCAN_TPB     = 256;
constexpr int SCAN_BLOCKS  = (int)(SITESc / SCAN_TPB); // 9600 (exact)

// ---------------------------------------------------------------------------
// CDNA5 async-to-LDS plumbing (guarded so the file always compiles).
// Builtin signature (probe-verified via diagnostics): (AS1 int*, AS3 int*, imm, imm)
// ---------------------------------------------------------------------------
#if defined(__AMDGCN__) && __has_builtin(__builtin_amdgcn_global_load_async_to_lds_b32)
#define ASYNC_LDS 1
typedef __attribute__((address_space(1))) int gas_int;  // global AS int
typedef __attribute__((address_space(3))) int las_int;  // LDS AS int
// Flat->AS1 is value-identity; flat LDS addr low 32 bits == LDS byte offset (ISA 10.2).
#define TO_GAS(p) ((gas_int*)(uintptr_t)(p))
#define TO_LAS(p) ((las_int*)(uintptr_t)(uint32_t)(uintptr_t)(p))
#endif

#if defined(__AMDGCN__) && __has_builtin(__builtin_amdgcn_s_wait_asynccnt)
#define WAIT_ASYNC_0() __builtin_amdgcn_s_wait_asynccnt(0)
#define WAIT_ASYNC_1() __builtin_amdgcn_s_wait_asynccnt(1)
#else
#define WAIT_ASYNC_0() asm volatile("s_wait_asynccnt 0x0" ::: "memory")
#define WAIT_ASYNC_1() asm volatile("s_wait_asynccnt 0x1" ::: "memory")
#endif

// Shared coordinate math: jnp.round == RNE (rintf), then clip.
__device__ __forceinline__ int site_of(float ex, float ey, long long e,
                                       const long long* __restrict__ offsets,
                                       int nOff) {
    int y = (int)rintf(ey * (float)Hc);
    int x = (int)rintf(ex * (float)Wc);
    y = min(Hc - 1, max(0, y));
    x = min(Wc - 1, max(0, x));
    int b = 0;
    for (int i = 0; i < nOff; ++i) b += (e >= offsets[i]) ? 1 : 0;
    return (b * Hc + y) * Wc + x;
}

// ---------------------------------------------------------------------------
// Kernel: 128-bit zero fill
// ---------------------------------------------------------------------------
__global__ void zero_f4_kernel(float4* __restrict__ p, long long n4) {
    long long i = (long long)blockIdx.x * blockDim.x + threadIdx.x;
    long long stride = (long long)gridDim.x * blockDim.x;
    float4 z = make_float4(0.f, 0.f, 0.f, 0.f);
    for (; i < n4; i += stride) p[i] = z;
}

// ============================ SORT-GATHER PATH =============================
// Pass A: per-event site + histogram (2M cheap u32 atomics, 2.45M bins)
__global__ void bin_kernel(const float* __restrict__ events,
                           const long long* __restrict__ offsets,
                           unsigned* __restrict__ sites,
                           unsigned* __restrict__ cnt,
                           long long N, int nOff) {
    long long e = (long long)blockIdx.x * blockDim.x + threadIdx.x;
    if (e >= N) return;
    int s = site_of(events[e * 2 + 0], events[e * 2 + 1], e, offsets, nOff);
    sites[e] = (unsigned)s;
    atomicAdd(&cnt[s], 1u);
}

// Pass B1: per-block exclusive scan; cursor[i] = in-block exclusive, bsum[blk]=total
__global__ void scan1_kernel(const unsigned* __restrict__ cnt,
                             unsigned* __restrict__ cursor,
                             unsigned* __restrict__ bsum) {
    __shared__ unsigned sh[SCAN_TPB];
    int t = threadIdx.x;
    long long i = (long long)blockIdx.x * SCAN_TPB + t;   // grid sized exactly
    unsigned v = cnt[i];
    sh[t] = v;
    __syncthreads();
    for (int off = 1; off < SCAN_TPB; off <<= 1) {
        unsigned add = (t >= off) ? sh[t - off] : 0u;
        __syncthreads();
        sh[t] += add;
        __syncthreads();
    }
    cursor[i] = sh[t] - v;                 // exclusive within block
    if (t == SCAN_TPB - 1) bsum[blockIdx.x] = sh[t];
}

// Pass B2: single-block sequential-chunk exclusive scan of the 9600 block sums
__global__ void scan2_kernel(unsigned* __restrict__ bsum, int nb) {
    __shared__ unsigned sh[SCAN_TPB];
    __shared__ unsigned carry;
    int t = threadIdx.x;
    if (t == 0) carry = 0u;
    __syncthreads();
    for (int base = 0; base < nb; base += SCAN_TPB) {
        int i = base + t;
        unsigned v = (i < nb) ? bsum[i] : 0u;
        sh[t] = v;
        __syncthreads();
        for (int off = 1; off < SCAN_TPB; off <<= 1) {
            unsigned add = (t >= off) ? sh[t - off] : 0u;
            __syncthreads();
            sh[t] += add;
            __syncthreads();
        }
        unsigned chunkTot = sh[SCAN_TPB - 1];
        unsigned myExcl = sh[t] - v + carry;
        if (i < nb) bsum[i] = myExcl;
        __syncthreads();
        if (t == 0) carry += chunkTot;
        __syncthreads();
    }
}

// Pass B3: add scanned block offsets -> cursor[i] = global exclusive start
__global__ void scan3_kernel(unsigned* __restrict__ cursor,
                             const unsigned* __restrict__ bsum) {
    long long i = (long long)blockIdx.x * SCAN_TPB + threadIdx.x;
    cursor[i] += bsum[blockIdx.x];
}

// Pass C: scatter event ids into order[]; afterwards cursor[s] == end(s)
__global__ void place_kernel(const unsigned* __restrict__ sites,
                             unsigned* __restrict__ cursor,
                             unsigned* __restrict__ order,
                             long long N) {
    long long e = (long long)blockIdx.x * blockDim.x + threadIdx.x;
    if (e >= N) return;
    unsigned s = sites[e];
    unsigned idx = atomicAdd(&cursor[s], 1u);
    order[idx] = (unsigned)e;
}

// Pass D: one wave32 per site; lane d = channel d. Coalesced 128B feature-row
// loads; writes every output site (zeros where empty) -> no zero-fill pass,
// no normalize pass, zero f32 atomics.
__global__ void gather_kernel(const float* __restrict__ feats,
                              const unsigned* __restrict__ order,
                              const unsigned* __restrict__ cursor, // == end
                              const unsigned* __restrict__ cnt,
                              float* __restrict__ out) {
    const int lane = threadIdx.x & 31;
    const int wid  = threadIdx.x >> 5;
    long long site = (long long)blockIdx.x * (blockDim.x >> 5) + wid;
    if (site >= SITESc) return;

    unsigned end = cursor[site];
    unsigned c   = cnt[site];
    unsigned beg = end - c;

    float acc = 0.f;
    for (unsigned k = beg; k < end; ++k) {
        unsigned ev = order[k];
        acc += feats[(long long)ev * DIMc + lane];
    }
    // true IEEE divide to match reference's feat_sum / max(counts, 1)
    float denom = (c > 1u) ? (float)c : 1.0f;
    float r = acc / denom;

    long long b  = site / HWc;
    long long hw = site - b * HWc;
    out[((long long)(b * DIMc + lane)) * HWc + hw] = r;
}

// ======================= FALLBACK ATOMIC PATH ==============================
// (used if ws_size is too small for the sort; keeps the async-LDS
//  double-buffered feed — the gfx1250 async path — in the binary)
__global__ void scatter_kernel(const float* __restrict__ events,
                               const float* __restrict__ feats,
                               const long long* __restrict__ offsets,
                               float* __restrict__ out,
                               float* __restrict__ cnt,
                               long long N, int nOff) {
    const int lane = threadIdx.x & 31;
    const int wid  = threadIdx.x >> 5;
    const long long wavesPerBlk = blockDim.x >> 5;
    const long long wave   = (long long)blockIdx.x * wavesPerBlk + wid;
    const long long nwaves = (long long)gridDim.x * wavesPerBlk;

#if ASYNC_LDS
    __shared__ float sbuf[8][2][DIMc];
    long long e = wave;
    if (e < N) {
        __builtin_amdgcn_global_load_async_to_lds_b32(
            TO_GAS(feats + e * DIMc + lane), TO_LAS(&sbuf[wid][0][lane]), 0, 0);
    }
    int parity = 0;
    for (; e < N; e += nwaves) {
        long long en = e + nwaves;
        if (en < N) {
            __builtin_amdgcn_global_load_async_to_lds_b32(
                TO_GAS(feats + en * DIMc + lane),
                TO_LAS(&sbuf[wid][parity ^ 1][lane]), 0, 0);
        }
        int s = site_of(events[e * 2 + 0], events[e * 2 + 1], e, offsets, nOff);
        long long b  = s / (int)HWc;
        long long hw = s - b * HWc;

        if (en < N) { WAIT_ASYNC_1(); } else { WAIT_ASYNC_0(); }
        asm volatile("" ::: "memory");
        const float v = *(volatile float*)&sbuf[wid][parity][lane];

        atomicAdd(&out[((long long)(b * DIMc + lane)) * HWc + hw], v);
        if (lane == 0) atomicAdd(&cnt[s], 1.0f);
        parity ^= 1;
    }
#else
    for (long long e = wave; e < N; e += nwaves) {
        int s = site_of(events[e * 2 + 0], events[e * 2 + 1], e, offsets, nOff);
        long long b  = s / (int)HWc;
        long long hw = s - b * HWc;
        const float v = feats[e * DIMc + lane];
        atomicAdd(&out[((long long)(b * DIMc + lane)) * HWc + hw], v);
        if (lane == 0) atomicAdd(&cnt[s], 1.0f);
    }
#endif
}

__global__ void normalize_kernel(float* __restrict__ out,
                                 const float* __restrict__ cnt) {
    long long site = (long long)blockIdx.x * blockDim.x + threadIdx.x;
    if (site >= SITESc) return;
    float c = cnt[site];
    if (c > 1.0f) {
        float inv = 1.0f / c;
        long long b  = site / HWc;
        long long hw = site - b * HWc;
        float* p = out + b * (long long)DIMc * HWc + hw;
#pragma unroll
        for (int d = 0; d < DIMc; ++d) p[(long long)d * HWc] *= inv;
    }
}

// ---------------------------------------------------------------------------
extern "C" void kernel_launch(void* const* d_in, const int* in_sizes, int n_in,
                              void* d_out, int out_size, void* d_ws, size_t ws_size,
                              hipStream_t stream) {
    const float*     events  = (const float*)d_in[0];      // [N, 2] f32
    const float*     feats   = (const float*)d_in[1];      // [N, 32] f32
    const long long* offsets = (const long long*)d_in[2];  // [8] i64 cumulative
    float* out = (float*)d_out;                            // [8, 32, 480, 640] f32

    const long long N    = (long long)in_sizes[0] / 2;
    const int       nOff = in_sizes[2];
    const int TPB = 256;

    // Workspace layout for sort-gather path (all offsets 64B-aligned)
    auto align64 = [](size_t x) { return (x + 63) & ~(size_t)63; };
    size_t off_cnt   = 0;
    size_t off_cur   = align64(off_cnt + (size_t)SITESc * 4);
    size_t off_sites = align64(off_cur + (size_t)SITESc * 4);
    size_t off_order = align64(off_sites + (size_t)N * 4);
    size_t off_bsum  = align64(off_order + (size_t)N * 4);
    size_t needed    = off_bsum + (size_t)SCAN_BLOCKS * 4;

    if (ws_size >= needed) {
        // ------------------- sort-gather path -------------------
        char* ws = (char*)d_ws;
        unsigned* cnt    = (unsigned*)(ws + off_cnt);
        unsigned* cursor = (unsigned*)(ws + off_cur);
        unsigned* sites  = (unsigned*)(ws + off_sites);
        unsigned* order  = (unsigned*)(ws + off_order);
        unsigned* bsum   = (unsigned*)(ws + off_bsum);

        long long n4c = SITESc * 4 / 16;
        zero_f4_kernel<<<(int)((n4c + TPB - 1) / TPB), TPB, 0, stream>>>(
            (float4*)cnt, n4c);

        int eblk = (int)((N + TPB - 1) / TPB);
        bin_kernel<<<eblk, TPB, 0, stream>>>(events, offsets, sites, cnt, N, nOff);

        scan1_kernel<<<SCAN_BLOCKS, SCAN_TPB, 0, stream>>>(cnt, cursor, bsum);
        scan2_kernel<<<1, SCAN_TPB, 0, stream>>>(bsum, SCAN_BLOCKS);
        scan3_kernel<<<SCAN_BLOCKS, SCAN_TPB, 0, stream>>>(cursor, bsum);

        place_kernel<<<eblk, TPB, 0, stream>>>(sites, cursor, order, N);

        int gblk = (int)(SITESc / (TPB / 32));  // 8 sites (waves) per block
        gather_kernel<<<gblk, TPB, 0, stream>>>(feats, order, cursor, cnt, out);
    } else {
        // ------------------- fallback atomic path -------------------
        float* cntf = (float*)d_ws;  // needs SITESc*4 bytes
        long long n4o = OUTNc / 4, n4c = SITESc / 4;
        zero_f4_kernel<<<(int)((n4o + TPB - 1) / TPB), TPB, 0, stream>>>(
            (float4*)out, n4o);
        zero_f4_kernel<<<(int)((n4c + TPB - 1) / TPB), TPB, 0, stream>>>(
            (float4*)cntf, n4c);
        scatter_kernel<<<8192, TPB, 0, stream>>>(events, feats, offsets,
                                                 out, cntf, N, nOff);
        normalize_kernel<<<(int)((SITESc + TPB - 1) / TPB), TPB, 0, stream>>>(
            out, cntf);
    }
}